// SQNRandLANet_19593640804570
// MI455X (gfx1250) — compile-verified
//
#include <hip/hip_runtime.h>
#include <hip/hip_bf16.h>
#include <stdint.h>

// Problem dims (fixed by the reference)
#define B_  4
#define C_  128
#define N_  16384
#define M_  4096
#define KP_ 16          // pool_idx K

typedef __attribute__((ext_vector_type(2))) float        v2f;
typedef __attribute__((ext_vector_type(8))) float        v8f;
typedef __attribute__((ext_vector_type(4))) unsigned int u32x4;
typedef __attribute__((ext_vector_type(8))) int          i32x8;
typedef __attribute__((ext_vector_type(4))) int          i32x4;

// ---------------------------------------------------------------------------
// Tensor Data Mover: 1-D contiguous copy of `nelem` f32 from global -> LDS.
// Descriptor per CDNA5 ISA ch.8 (group0: count=1,type=2; group1: data_size=4B,
// tensor_dim0=tile_dim0=nelem, tensor_dim1=1, stride0=nelem).
// 6-arg builtin form (clang-23 / therock-10.0 headers).
// ---------------------------------------------------------------------------
__device__ __forceinline__ void tdm_load_f32(unsigned lds_addr, const void* gptr, unsigned nelem) {
  unsigned long long ga = (unsigned long long)gptr;
  u32x4 g0;
  g0[0] = 1u;                                            // count=1, is_restore=0, gather off
  g0[1] = lds_addr;                                      // LDS byte address
  g0[2] = (unsigned)(ga & 0xFFFFFFFFu);                  // global_addr[31:0]
  g0[3] = (unsigned)((ga >> 32) & 0x01FFFFFFu) | (2u << 30); // global_addr[56:32], type=2
  i32x8 g1;
  g1[0] = (int)(2u << 16);                               // workgroup_mask=0, data_size=2 (4B)
  g1[1] = (int)((nelem & 0xFFFFu) << 16);                // tensor_dim0[15:0]
  g1[2] = (int)(((nelem >> 16) & 0xFFFFu) | (1u << 16)); // tensor_dim0[31:16], tensor_dim1=1
  g1[3] = (int)((nelem & 0xFFFFu) << 16);                // tensor_dim1[31:16]=0, tile_dim0=nelem
  g1[4] = 0;                                             // tile_dim1=0, tile_dim2=0
  g1[5] = (int)nelem;                                    // tensor_dim0_stride[31:0]
  g1[6] = 0;
  g1[7] = 0;
  i32x4 z4 = {0, 0, 0, 0};
  i32x8 z8 = {0, 0, 0, 0, 0, 0, 0, 0};
  __builtin_amdgcn_tensor_load_to_lds(g0, g1, z4, z4, z8, 0); // groups 2/3 unused, cpol=0
}

// Branchless sorted top-3 (ascending) insert with index tracking.
__device__ __forceinline__ void top3_insert(float d, int i,
    float& d0, float& d1, float& d2, int& i0, int& i1, int& i2) {
  bool lt0 = d < d0, lt1 = d < d1, lt2 = d < d2;
  float nd2 = lt1 ? d1 : (lt2 ? d : d2); int ni2 = lt1 ? i1 : (lt2 ? i : i2);
  float nd1 = lt0 ? d0 : (lt1 ? d : d1); int ni1 = lt0 ? i0 : (lt1 ? i : i1);
  d0 = lt0 ? d : d0;                      i0 = lt0 ? i : i0;
  d1 = nd1; i1 = ni1;
  d2 = nd2; i2 = ni2;
}

// ---------------------------------------------------------------------------
// Prep: pack supports as (x, y, z, |s|^2) float4 for the WMMA A-operand.
// ---------------------------------------------------------------------------
__global__ void prep_support(const float* __restrict__ xyz_s, float* __restrict__ sup) {
  int t = blockIdx.x * blockDim.x + threadIdx.x;
  if (t >= B_ * M_) return;
  float x = xyz_s[3 * t + 0], y = xyz_s[3 * t + 1], z = xyz_s[3 * t + 2];
  float4 v = make_float4(x, y, z, x * x + y * y + z * z);
  ((float4*)sup)[t] = v;
}

// Prep: pack queries as (-2x, -2y, -2z, 1) for the WMMA B-operand, plus |q|^2.
__global__ void prep_query(const float* __restrict__ xyz_q,
                           float* __restrict__ qp, float* __restrict__ qn) {
  int t = blockIdx.x * blockDim.x + threadIdx.x;
  if (t >= B_ * N_) return;
  float x = xyz_q[3 * t + 0], y = xyz_q[3 * t + 1], z = xyz_q[3 * t + 2];
  float4 v = make_float4(-2.f * x, -2.f * y, -2.f * z, 1.f);
  ((float4*)qp)[t] = v;
  qn[t] = x * x + y * y + z * z;
}

// ---------------------------------------------------------------------------
// Gather-max-pool: pooled[b,c,m] = max_k feature[b,c,idx[b,m,k]]
// One block per (b,m); 128 threads = all channels; indices staged in LDS.
// ---------------------------------------------------------------------------
__global__ __launch_bounds__(128)
void pool_max(const float* __restrict__ feat, const int* __restrict__ idx,
              float* __restrict__ pooled) {
  __shared__ int sidx[KP_];
  int b = blockIdx.y, m = blockIdx.x, c = threadIdx.x;
  if (threadIdx.x < KP_) sidx[threadIdx.x] = idx[(b * M_ + m) * KP_ + threadIdx.x];
  __syncthreads();
  const float* f = feat + (size_t)(b * C_ + c) * N_;
  float v = f[sidx[0]];
#pragma unroll
  for (int k = 1; k < KP_; ++k) v = fmaxf(v, f[sidx[k]]);
  pooled[(size_t)(b * C_ + c) * M_ + m] = v;
}

// ---------------------------------------------------------------------------
// Main kernel: WMMA distance tiles + top-3 + 3-NN weighted interpolation.
// Block = 128 threads (4 waves); each wave owns a 16-query tile (64/block).
// Supports stream through LDS in double-buffered 1024-support (16KB) chunks
// via the Tensor Data Mover; each chunk is scanned with v_wmma_f32_16x16x4_f32.
// ---------------------------------------------------------------------------
#define CHUNK   1024
#define NCHUNKS (M_ / CHUNK)
#define TILES   (CHUNK / 16)

__global__ __launch_bounds__(128)
void knn_interp(const float* __restrict__ sup_packed,
                const float* __restrict__ q_packed,
                const float* __restrict__ qnorm,
                const float* __restrict__ pooled,
                float* __restrict__ out) {
  extern __shared__ char smem[];
  float* sbuf = (float*)smem;                          // 2 * CHUNK * 4 f32 (32KB)
  float* swgt = (float*)(smem + 2 * CHUNK * 16);       // 64*3 weights
  int*   sknn = (int*)(swgt + 64 * 3);                 // 64*3 indices

  const int b    = blockIdx.y;
  const int nb0  = blockIdx.x * 64;
  const int wave = threadIdx.x >> 5;
  const int lane = threadIdx.x & 31;
  const int qcol = lane & 15;     // query column within 16-wide tile
  const int half = lane >> 4;     // 0: K=0,1 / rows 0-7   1: K=2,3 / rows 8-15

  // B-operand (constant over the whole scan): queries for this wave's tile.
  const int n0 = nb0 + wave * 16;
  const float* qptr = q_packed + ((size_t)(b * N_ + n0 + qcol) << 2) + half * 2;
  v2f breg; breg.x = qptr[0]; breg.y = qptr[1];

  const unsigned lds_base = (unsigned)(unsigned long long)(void*)sbuf; // LDS byte offset
  const float* gsup = sup_packed + (size_t)b * M_ * 4;

  if (wave == 0) tdm_load_f32(lds_base, gsup, CHUNK * 4);   // prime chunk 0

  float d0 = 3.4e38f, d1 = 3.4e38f, d2 = 3.4e38f;
  int   i0 = 0, i1 = 0, i2 = 0;

  for (int ch = 0; ch < NCHUNKS; ++ch) {
    if (wave == 0) {
      if (ch + 1 < NCHUNKS)
        tdm_load_f32(lds_base + ((ch + 1) & 1) * (CHUNK * 16),
                     gsup + (size_t)(ch + 1) * CHUNK * 4, CHUNK * 4);
      // TENSORcnt is in-order per wave: <=1 completes the older (current) chunk.
      if (ch + 1 < NCHUNKS) __builtin_amdgcn_s_wait_tensorcnt(1);
      else                  __builtin_amdgcn_s_wait_tensorcnt(0);
    }
    __syncthreads();                                   // chunk `ch` visible to all waves

    const float* sb = sbuf + (ch & 1) * (CHUNK * 4);
    const int mchunk = ch * CHUNK;
#pragma unroll 2
    for (int t = 0; t < TILES; ++t) {
      // A-operand: 16 supports x (x,y,z,|s|^2); lane layout per ISA 16x4 f32.
      const float* ap = sb + (((t << 4) + qcol) << 2) + half * 2;
      v2f areg; areg.x = ap[0]; areg.y = ap[1];
      v8f acc = {0.f, 0.f, 0.f, 0.f, 0.f, 0.f, 0.f, 0.f};
      // D[m,q] = -2 q.s + |s|^2  (== dist - |q|^2; row-constant |q|^2 added later)
      acc = __builtin_amdgcn_wmma_f32_16x16x4_f32(
          false, areg, false, breg, (short)0, acc, false, false);
      const int mrow = mchunk + (t << 4) + (half << 3);
#pragma unroll
      for (int r = 0; r < 8; ++r)
        top3_insert(acc[r], mrow + r, d0, d1, d2, i0, i1, i2);
    }
    __syncthreads();                                   // done reading before TDM reuses buffer
  }

  // Lanes l and l^16 hold the same query column: merge their top-3 lists.
  {
    float od0 = __shfl_xor(d0, 16), od1 = __shfl_xor(d1, 16), od2 = __shfl_xor(d2, 16);
    int   oi0 = __shfl_xor(i0, 16), oi1 = __shfl_xor(i1, 16), oi2 = __shfl_xor(i2, 16);
    top3_insert(od0, oi0, d0, d1, d2, i0, i1, i2);
    top3_insert(od1, oi1, d0, d1, d2, i0, i1, i2);
    top3_insert(od2, oi2, d0, d1, d2, i0, i1, i2);
  }

  if (half == 0) {
    const int n  = n0 + qcol;
    const float qn = qnorm[b * N_ + n];
    const float r0 = 1.f / (d0 + qn + 1e-8f);
    const float r1 = 1.f / (d1 + qn + 1e-8f);
    const float r2 = 1.f / (d2 + qn + 1e-8f);
    const float inv = 1.f / (r0 + r1 + r2);
    const int slot = wave * 16 + qcol;
    swgt[slot * 3 + 0] = r0 * inv;
    swgt[slot * 3 + 1] = r1 * inv;
    swgt[slot * 3 + 2] = r2 * inv;
    sknn[slot * 3 + 0] = i0;
    sknn[slot * 3 + 1] = i1;
    sknn[slot * 3 + 2] = i2;
  }
  __syncthreads();

  // Output: out[b,c,nb0+q] = sum_k w_k * pooled[b,c,knn_k].
  // Threads map q = tid&63 (n-contiguous, coalesced stores), c strided by 2.
  const int qoff = threadIdx.x & 63;
  const int c0   = threadIdx.x >> 6;
  const float w0 = swgt[qoff * 3 + 0], w1 = swgt[qoff * 3 + 1], w2 = swgt[qoff * 3 + 2];
  const int   j0 = sknn[qoff * 3 + 0], j1 = sknn[qoff * 3 + 1], j2 = sknn[qoff * 3 + 2];
  const float* pb = pooled + (size_t)b * C_ * M_;
  float* ob = out + (size_t)b * C_ * N_ + nb0 + qoff;
#pragma unroll 4
  for (int c = c0; c < C_; c += 2) {
    const float* pc = pb + (size_t)c * M_;
    ob[(size_t)c * N_] = w0 * pc[j0] + w1 * pc[j1] + w2 * pc[j2];
  }
}

// ---------------------------------------------------------------------------
extern "C" void kernel_launch(void* const* d_in, const int* in_sizes, int n_in,
                              void* d_out, int out_size, void* d_ws, size_t ws_size,
                              hipStream_t stream) {
  const float* feature  = (const float*)d_in[0];   // (B,C,N,1)
  const int*   pool_idx = (const int*)d_in[1];     // (B,M,K)
  const float* xyz_q    = (const float*)d_in[2];   // (B,N,3)
  const float* xyz_s    = (const float*)d_in[3];   // (B,M,3)
  float* out = (float*)d_out;                      // (B,C,N)

  float* ws     = (float*)d_ws;
  float* pooled = ws;                                   // B*C*M
  float* sup    = pooled + (size_t)B_ * C_ * M_;        // B*M*4
  float* qp     = sup    + (size_t)B_ * M_ * 4;         // B*N*4
  float* qn     = qp     + (size_t)B_ * N_ * 4;         // B*N

  prep_support<<<(B_ * M_ + 255) / 256, 256, 0, stream>>>(xyz_s, sup);
  prep_query<<<(B_ * N_ + 255) / 256, 256, 0, stream>>>(xyz_q, qp, qn);
  pool_max<<<dim3(M_, B_), 128, 0, stream>>>(feature, pool_idx, pooled);

  const size_t lds_bytes = 2 * CHUNK * 16 + 64 * 3 * sizeof(float) + 64 * 3 * sizeof(int);
  knn_interp<<<dim3(N_ / 64, B_), 128, lds_bytes, stream>>>(sup, qp, qn, pooled, out);
}